// RelationRepLayer_43147241455989
// MI455X (gfx1250) — compile-verified
//
#include <hip/hip_runtime.h>
#include <hip/hip_bf16.h>
#include <cstdint>

// ---------------------------------------------------------------------------
// RelationRepLayer on gfx1250 (MI455X), fp32 end-to-end using
// V_WMMA_F32_16X16X4_F32 (full-precision CDNA5 matrix op).
//
// Shapes: B=2, S=512, K=16, H=256, FFN_IN=768, FFN_MID=1152, M=B*K*K=512.
//
// Pipeline:
//   1) pack W1,W2 into K-pair-major layout  (b64-loadable B fragments)
//   2) build packed 512x768 activation matrix (head|tail|maxpool-context)
//   3) hidden = relu(A @ W1 + b1)  via f32 WMMA
//   4) out    = hidden @ W2 + b2   via f32 WMMA  (row-major == (B,K,K,H))
// ---------------------------------------------------------------------------

#define Hdim     256
#define Sdim     512
#define Kspan    16
#define Bdim     2
#define Mrows    (Bdim * Kspan * Kspan)   // 512
#define FFN_IN   (3 * Hdim)               // 768
#define FFN_MID  1152

typedef __attribute__((ext_vector_type(2))) float v2f;
typedef __attribute__((ext_vector_type(8))) float v8f;

// ---------------------------------------------------------------------------
// Pack W (Kd x N, row-major) into Wp[q][n][2] = { W[2q][n], W[2q+1][n] }.
// Makes each WMMA B-fragment one contiguous 8-byte load.
// ---------------------------------------------------------------------------
template<int N>
__global__ void pack_w_kernel(const float* __restrict__ W,
                              float*       __restrict__ Wp,
                              int total /* = (Kd/2)*N */) {
    int idx = blockIdx.x * 256 + threadIdx.x;
    if (idx >= total) return;
    int q = idx / N;
    int n = idx - q * N;
    v2f p;
    p.x = W[(2 * q)     * N + n];
    p.y = W[(2 * q + 1) * N + n];
    *(v2f*)(Wp + (size_t)idx * 2) = p;
}

// ---------------------------------------------------------------------------
// Kernel: build packed activation matrix relA (512 x 768):
//   [0:256) head rep | [256:512) tail rep | [512:768) context (or fallback)
// One block per (b,i,j) pair; one thread per channel h.
// ---------------------------------------------------------------------------
__global__ void build_rel_reps_kernel(
    const float*   __restrict__ token_reps,   // (B,S,H)
    const uint8_t* __restrict__ token_masks,  // (B,S) bool
    const uint8_t* __restrict__ rel_masks,    // (B,K,K) bool
    const int*     __restrict__ span_ids,     // (B,K,2)
    const float*   __restrict__ span_reps,    // (B,K,H)
    const float*   __restrict__ no_ctx,       // (H,)
    float*         __restrict__ relA)         // (512,768)
{
    const int p = blockIdx.x;
    const int b = p >> 8;
    const int i = (p >> 4) & 15;
    const int j = p & 15;
    const int h = threadIdx.x;

    const int hs = span_ids[(b * Kspan + i) * 2 + 0];
    const int he = span_ids[(b * Kspan + i) * 2 + 1];
    const int ts = span_ids[(b * Kspan + j) * 2 + 0];
    const int te = span_ids[(b * Kspan + j) * 2 + 1];

    const int min_end   = (he < te) ? he : te;
    const int max_start = (hs > ts) ? hs : ts;
    const bool valid = (min_end < max_start) &&
                       (rel_masks[(b * Kspan + i) * Kspan + j] != 0);

    float m = -6.5e4f;                 // NEG_LIMIT
    bool any = false;
    if (valid) {
        const float*   trow = token_reps + (size_t)b * Sdim * Hdim;
        const uint8_t* tm   = token_masks + (size_t)b * Sdim;
        for (int s = min_end; s < max_start; ++s) {
            if (tm[s]) {
                any = true;
                m = fmaxf(m, trow[(size_t)s * Hdim + h]);
            }
        }
    }
    const float ctx = any ? m : no_ctx[h];

    float* row = relA + (size_t)p * FFN_IN;
    row[h]            = span_reps[((size_t)b * Kspan + i) * Hdim + h];
    row[Hdim + h]     = span_reps[((size_t)b * Kspan + j) * Hdim + h];
    row[2 * Hdim + h] = ctx;
}

// ---------------------------------------------------------------------------
// fp32 WMMA GEMM on packed weights:
//   D(MxN) = act( A(MxKd) @ W(KdxN) + bias ),  W given as Wp (K-pair-major).
// Each wave: one 16x64 strip (A frag reused across 4 B frags, 4 WMMAs/K-step).
// N, Kd compile-time so all strides fold into immediate offsets.
// Grid must give exactly (M/16)*(N/64) waves (EXEC all-ones at every WMMA).
// ---------------------------------------------------------------------------
template<int N, int Kd, bool RELU>
__global__ void wmma_gemm_f32_kernel(
    const float* __restrict__ A,      // (M, Kd) row-major
    const float* __restrict__ Wp,     // packed: (Kd/2, N, 2)
    const float* __restrict__ bias,   // (N,)
    float*       __restrict__ D)      // (M, N) row-major
{
    constexpr int STRIPS = N >> 6;                 // N / 64
    const int wave  = blockIdx.x * (blockDim.x >> 5) + (threadIdx.x >> 5);
    const int tileM = wave / STRIPS;
    const int n0    = (wave % STRIPS) << 6;

    const int lane = threadIdx.x & 31;
    const int m    = lane & 15;
    const int half = lane >> 4;
    const int kB   = half << 1;                    // {0, 2}

    v8f acc0 = {}, acc1 = {}, acc2 = {}, acc3 = {};

    const float* Ap = A  + (tileM * 16 + m) * Kd + kB;
    const float* Bp = Wp + (half * N + n0 + m) * 2;   // q = k/2 + half

    #pragma unroll 4
    for (int k = 0; k < Kd; k += 4) {
        v2f a  = *(const v2f*)Ap;            Ap += 4;
        v2f b0 = *(const v2f*)(Bp + 0);
        v2f b1 = *(const v2f*)(Bp + 32);
        v2f b2 = *(const v2f*)(Bp + 64);
        v2f b3 = *(const v2f*)(Bp + 96);
        Bp += 4 * N;                         // q advances by 2 rows of N*2

        acc0 = __builtin_amdgcn_wmma_f32_16x16x4_f32(false, a, false, b0,
                                                     (short)0, acc0, false, false);
        acc1 = __builtin_amdgcn_wmma_f32_16x16x4_f32(false, a, false, b1,
                                                     (short)0, acc1, false, false);
        acc2 = __builtin_amdgcn_wmma_f32_16x16x4_f32(false, a, false, b2,
                                                     (short)0, acc2, false, false);
        acc3 = __builtin_amdgcn_wmma_f32_16x16x4_f32(false, a, false, b3,
                                                     (short)0, acc3, false, false);
    }

    // Epilogue: d[g] -> D[tileM*16 + g + 8*half][n0 + t*16 + m]
    const float bb0 = bias[n0 + m];
    const float bb1 = bias[n0 + 16 + m];
    const float bb2 = bias[n0 + 32 + m];
    const float bb3 = bias[n0 + 48 + m];
    float* drow = D + (tileM * 16 + half * 8) * N + n0 + m;
    #pragma unroll
    for (int g = 0; g < 8; ++g) {
        float v0 = acc0[g] + bb0;
        float v1 = acc1[g] + bb1;
        float v2 = acc2[g] + bb2;
        float v3 = acc3[g] + bb3;
        if (RELU) {
            v0 = fmaxf(v0, 0.0f); v1 = fmaxf(v1, 0.0f);
            v2 = fmaxf(v2, 0.0f); v3 = fmaxf(v3, 0.0f);
        }
        drow[0]  = v0;
        drow[16] = v1;
        drow[32] = v2;
        drow[48] = v3;
        drow += N;
    }
}

// ---------------------------------------------------------------------------
// Host-side launch. Inputs (setup_inputs order):
//  0 token_reps (B,S,H) f32       5 W1 (768,1152) f32
//  1 token_masks (B,S) bool       6 b1 (1152,) f32
//  2 rel_masks (B,K,K) bool       7 W2 (1152,256) f32
//  3 cand_span_ids (B,K,2) i32    8 b2 (256,) f32
//  4 cand_span_reps (B,K,H) f32   9 no_context_embedding (256,) f32
// Output: (B,K,K,H) f32 == row-major (512,256)
// ---------------------------------------------------------------------------
extern "C" void kernel_launch(void* const* d_in, const int* in_sizes, int n_in,
                              void* d_out, int out_size, void* d_ws, size_t ws_size,
                              hipStream_t stream) {
    const float*   token_reps  = (const float*)d_in[0];
    const uint8_t* token_masks = (const uint8_t*)d_in[1];
    const uint8_t* rel_masks   = (const uint8_t*)d_in[2];
    const int*     span_ids    = (const int*)d_in[3];
    const float*   span_reps   = (const float*)d_in[4];
    const float*   W1          = (const float*)d_in[5];
    const float*   b1          = (const float*)d_in[6];
    const float*   W2          = (const float*)d_in[7];
    const float*   b2          = (const float*)d_in[8];
    const float*   no_ctx      = (const float*)d_in[9];
    float*         out         = (float*)d_out;

    // Workspace layout (floats):
    //   relA   : 512*768            = 393216
    //   hidden : 512*1152           = 589824
    //   Wp1    : 768*1152 (packed)  = 884736
    //   Wp2    : 1152*256 (packed)  = 294912
    float* relA   = (float*)d_ws;
    float* hidden = relA   + (size_t)Mrows * FFN_IN;
    float* Wp1    = hidden + (size_t)Mrows * FFN_MID;
    float* Wp2    = Wp1    + (size_t)FFN_IN * FFN_MID;

    // Stage 0: pack weights into K-pair-major for b64 B-fragment loads.
    {
        const int tot1 = (FFN_IN  / 2) * FFN_MID;   // 442368
        const int tot2 = (FFN_MID / 2) * Hdim;      // 147456
        pack_w_kernel<FFN_MID><<<(tot1 + 255) / 256, 256, 0, stream>>>(W1, Wp1, tot1);
        pack_w_kernel<Hdim>   <<<(tot2 + 255) / 256, 256, 0, stream>>>(W2, Wp2, tot2);
    }

    // Stage 1: build packed (512 x 768) activation matrix.
    build_rel_reps_kernel<<<Mrows, Hdim, 0, stream>>>(
        token_reps, token_masks, rel_masks, span_ids, span_reps, no_ctx, relA);

    // Stage 2: hidden = relu(relA @ W1 + b1)   (512x768 @ 768x1152)
    // waves = (512/16)*(1152/64) = 576 -> 72 blocks x 8 waves.
    wmma_gemm_f32_kernel<FFN_MID, FFN_IN, true><<<72, 256, 0, stream>>>(
        relA, Wp1, b1, hidden);

    // Stage 3: out = hidden @ W2 + b2          (512x1152 @ 1152x256)
    // waves = (512/16)*(256/64) = 128 -> 16 blocks x 8 waves.
    wmma_gemm_f32_kernel<Hdim, FFN_MID, false><<<16, 256, 0, stream>>>(
        hidden, Wp2, b2, out);
}